// HotAnchorLayer_32830730011548
// MI455X (gfx1250) — compile-verified
//
#include <hip/hip_runtime.h>
#include <hip/hip_bf16.h>

typedef __attribute__((ext_vector_type(2))) float v2f;
typedef __attribute__((ext_vector_type(8))) float v8f;

#define COUNTS 1000
#define C_CH 256
#define NB 4

// ---------------------------------------------------------------------------
// Kernel 1: per-(b,c) mean over H*W.  One block per (b*C + c), float4 loads.
// ---------------------------------------------------------------------------
__global__ void hot_mean_kernel(const float* __restrict__ x,
                                float* __restrict__ mean, int HW) {
    const int tid = threadIdx.x;
    const float4* p = (const float4*)(x + (size_t)blockIdx.x * HW);
    const int n4 = HW >> 2;
    float s = 0.0f;
    for (int i = tid; i < n4; i += 256) {
        float4 v = p[i];
        s += (v.x + v.y) + (v.z + v.w);
    }
    __shared__ float red[256];
    red[tid] = s;
    __syncthreads();
    for (int off = 128; off > 0; off >>= 1) {
        if (tid < off) red[tid] += red[tid + off];
        __syncthreads();
    }
    if (tid == 0) mean[blockIdx.x] = red[0] / (float)HW;
}

// ---------------------------------------------------------------------------
// Kernel 2: heat[b,hw] = sum_c |x[b,c,hw] - mean[b,c]| via f32 WMMA.
// Each wave owns 16 consecutive pixels.  A (16x4 f32) = |diff| per documented
// layout (lane m: M=m, K={0,1} in VGPR0/1; lane 16+m: K={2,3}); B = ones.
// Channel loop is strength-reduced (running pointer, no u64 muls) and
// unrolled 4x over two alternating accumulators to decouple the WMMA C-chain
// from the load latency.  D-accumulation is linear, so heat = D0 + D1.
// ---------------------------------------------------------------------------
__global__ void hot_heat_wmma_kernel(const float* __restrict__ x,
                                     const float* __restrict__ mean,
                                     float* __restrict__ heat, int HW) {
    __shared__ float smean[C_CH];
    const int tid = threadIdx.x;
    const int b = blockIdx.y;
    smean[tid] = mean[b * C_CH + tid];
    __syncthreads();

    const int wave = tid >> 5;
    const int lane = tid & 31;
    const int m = lane & 15;       // pixel within tile -> A row M
    const int khalf = lane >> 4;   // 0: K={0,1}  1: K={2,3}
    const int tileBase = blockIdx.x * 128 + wave * 16;

    const size_t chStep = (size_t)4 * HW;  // elements between channel quads
    const float* p = x + (size_t)b * C_CH * HW + (size_t)(tileBase + m)
                     + (size_t)(khalf << 1) * HW;
    int cA = khalf << 1;

    v8f acc0 = {};
    v8f acc1 = {};
    v2f ones;
    ones.x = 1.0f;
    ones.y = 1.0f;

    for (int c0 = 0; c0 < C_CH; c0 += 16) {
        if (c0 + 16 < C_CH)
            __builtin_prefetch(p + (size_t)16 * HW, 0, 0);
        #pragma unroll
        for (int u = 0; u < 4; ++u) {
            const float v0 = __builtin_fabsf(p[0] - smean[cA]);
            const float v1 = __builtin_fabsf(p[HW] - smean[cA + 1]);
            v2f a;
            a.x = v0;  // K = 2*khalf     (VGPR 0)
            a.y = v1;  // K = 2*khalf + 1 (VGPR 1)
            if (u & 1)
                acc1 = __builtin_amdgcn_wmma_f32_16x16x4_f32(
                    false, a, false, ones, (short)0, acc1, false, false);
            else
                acc0 = __builtin_amdgcn_wmma_f32_16x16x4_f32(
                    false, a, false, ones, (short)0, acc0, false, false);
            p += chStep;
            cA += 4;
        }
    }

    if (m == 0) {  // lanes 0 (M=0..7) and 16 (M=8..15) write results
        float* hp = heat + (size_t)b * HW + tileBase + khalf * 8;
        #pragma unroll
        for (int r = 0; r < 8; ++r) hp[r] = acc0[r] + acc1[r];
    }
}

// ---------------------------------------------------------------------------
// Kernel 3: exact k-th largest value per batch via 4-pass MSB-first radix
// select on float bit patterns (heat >= 0 so uint order == float order).
// ---------------------------------------------------------------------------
__global__ void hot_kth_kernel(const float* __restrict__ heat,
                               unsigned* __restrict__ kth, int HW) {
    const int tid = threadIdx.x;
    const float* h = heat + (size_t)blockIdx.x * HW;
    __shared__ unsigned hist[256];
    __shared__ unsigned s_prefix;
    __shared__ int s_remain;

    unsigned prefix = 0;
    int remain = COUNTS;

    for (int pass = 3; pass >= 0; --pass) {
        const int shift = pass * 8;
        const unsigned hiMask = (pass == 3) ? 0u : (0xFFFFFFFFu << (shift + 8));
        hist[tid] = 0;
        __syncthreads();
        for (int i = tid; i < HW; i += 256) {
            const unsigned key = __float_as_uint(h[i]);
            if ((key & hiMask) == prefix)
                atomicAdd(&hist[(key >> shift) & 0xFFu], 1u);
        }
        __syncthreads();
        if (tid == 0) {
            int r = remain;
            unsigned sel = 0;
            for (int bkt = 255; bkt >= 0; --bkt) {
                const int c = (int)hist[bkt];
                if (r - c <= 0) { sel = (unsigned)bkt; break; }
                r -= c;
            }
            s_prefix = prefix | (sel << shift);
            s_remain = r;
        }
        __syncthreads();
        prefix = s_prefix;
        remain = s_remain;
        __syncthreads();
    }
    if (tid == 0) kth[blockIdx.x] = prefix;
}

// ---------------------------------------------------------------------------
// Kernel 4: take the first COUNTS flat indices (ascending) with heat >= kth
// (== stable argsort of the 0/1 mask in the reference) and emit the 3 boxes
// per center directly into the output.
// ---------------------------------------------------------------------------
__global__ void hot_select_boxes_kernel(const float* __restrict__ heat,
                                        const unsigned* __restrict__ kthBits,
                                        float* __restrict__ out,
                                        int HW, int W, float invStride,
                                        float scale, int rowOff) {
    const int tid = threadIdx.x;
    const int b = blockIdx.x;
    const float* h = heat + (size_t)b * HW;
    const float kth = __uint_as_float(kthBits[b]);

    __shared__ int psum[256];
    __shared__ int s_base;
    if (tid == 0) s_base = 0;
    __syncthreads();

    const float ratios[3] = {0.5f, 1.0f, 2.0f};

    for (int chunk = 0; chunk < HW; chunk += 256) {  // HW is a multiple of 256
        const int i = chunk + tid;
        const int msk = (h[i] >= kth) ? 1 : 0;
        psum[tid] = msk;
        __syncthreads();
        for (int off = 1; off < 256; off <<= 1) {
            const int v = (tid >= off) ? psum[tid - off] : 0;
            __syncthreads();
            psum[tid] += v;
            __syncthreads();
        }
        const int pos = s_base + psum[tid] - msk;
        const int total = psum[255];

        if (msk && pos < COUNTS) {
            const float py = (float)(i / W);
            const float px = (float)(i % W);
            const float cy = py * invStride;  // == py / (1024/H), exact (pow2)
            const float cx = px * invStride;
            #pragma unroll
            for (int a = 0; a < 3; ++a) {
                const float sq = sqrtf(ratios[a]);
                const float hh = scale / sq;
                const float ww = scale * sq;
                const float y1 = fminf(fmaxf((cy - 0.5f * hh) * (1.0f / 1024.0f), 0.0f), 1.0f);
                const float x1 = fminf(fmaxf((cx - 0.5f * ww) * (1.0f / 1024.0f), 0.0f), 1.0f);
                const float y2 = fminf(fmaxf((cy + 0.5f * hh) * (1.0f / 1024.0f), 0.0f), 1.0f);
                const float x2 = fminf(fmaxf((cx + 0.5f * ww) * (1.0f / 1024.0f), 0.0f), 1.0f);
                float* o = out + ((size_t)b * 12000 + rowOff + (size_t)pos * 3 + a) * 4;
                o[0] = y1; o[1] = x1; o[2] = y2; o[3] = x2;
            }
        }
        __syncthreads();
        if (tid == 0) s_base += total;
        __syncthreads();
        if (s_base >= COUNTS) break;  // uniform, deterministic early exit
    }
}

// ---------------------------------------------------------------------------
extern "C" void kernel_launch(void* const* d_in, const int* in_sizes, int n_in,
                              void* d_out, int out_size, void* d_ws, size_t ws_size,
                              hipStream_t stream) {
    (void)in_sizes; (void)n_in; (void)out_size; (void)ws_size;

    const float* fmaps[4] = {
        (const float*)d_in[0], (const float*)d_in[1],
        (const float*)d_in[2], (const float*)d_in[3]};
    const int HWs[4] = {256 * 256, 128 * 128, 64 * 64, 32 * 32};
    const int Ws[4] = {256, 128, 64, 32};
    const float scales[4] = {32.0f, 64.0f, 128.0f, 256.0f};

    float* out = (float*)d_out;
    char* ws = (char*)d_ws;
    float* heat = (float*)ws;                          // <= 4*65536 f32 = 1 MB, reused per level
    float* mean = (float*)(ws + (1u << 20));           // 1024 f32
    unsigned* kth = (unsigned*)(ws + (1u << 20) + 4096);  // 4 u32

    for (int L = 0; L < 4; ++L) {
        const int HW = HWs[L];
        hot_mean_kernel<<<NB * C_CH, 256, 0, stream>>>(fmaps[L], mean, HW);

        dim3 g(HW / 128, NB);
        hot_heat_wmma_kernel<<<g, 256, 0, stream>>>(fmaps[L], mean, heat, HW);

        hot_kth_kernel<<<NB, 256, 0, stream>>>(heat, kth, HW);

        const float invStride = (float)Ws[L] / 1024.0f;  // centers = p / (1024/H)
        hot_select_boxes_kernel<<<NB, 256, 0, stream>>>(heat, kth, out, HW, Ws[L],
                                                        invStride, scales[L], L * 3000);
    }
}